// rkhs_ouroboros_63462436766291
// MI455X (gfx1250) — compile-verified
//
#include <hip/hip_runtime.h>
#include <hip/hip_bf16.h>

// ---------------------------------------------------------------------------
// RKHS-Ouroboros for MI455X (gfx1250, wave32).
//
//   build_input      : X0[b,s,:] = mirrored (x, dxdt*TAU/dt) sequence, S = 2L
//   per (mamba m, layer l):
//     mamba_phase1   : per-chunk local scan -> carry (Aprod, Bsum) per lane
//     mamba_phase2   : serial scan of chunk carries (32 lanes = ED x N)
//     mamba_phase3   : re-run chunk with incoming prefix; per-lane h*C staged
//                      to LDS, reduction/gate/out_proj done in a parallel pass
//                      (NO cross-lane ops inside the serial scan loop)
//   final_kernel     : WMMA f32 16x16x4 for weights = control @ kw^T + b,
//                      then phi/omega/gamma/yhat.
// ---------------------------------------------------------------------------

#define WAVE 32
#define BLK  64                               // 2 waves: wave0 scans, both stage/reduce
static constexpr int   B_    = 2;             // batch
static constexpr int   NS_   = 16;            // D_STATE
static constexpr int   TCH   = 512;           // chunk length (S % TCH == 0)
static constexpr int   CST   = 36;            // contrib LDS row stride (pad: 36t%64 spreads banks)
static constexpr float TAU_  = 1.0f / 10000.0f;

struct LayerParams {            // jax-pytree (alphabetical) leaf order per layer:
  const float* A_log;           // (ED,16)
  const float* Dp;              // (ED,)
  const float* conv_b;          // (ED,)
  const float* conv_w;          // (4,ED)
  const float* dt_b;            // (ED,)
  const float* dt_w;            // (ED,1)
  const float* in_proj;         // (4,2)
  const float* norm_w;          // (2,)
  const float* out_proj;        // (2,2)
  const float* x_proj;          // (33,2)
};

__device__ __forceinline__ float siluf(float v)     { return v / (1.f + __expf(-v)); }
__device__ __forceinline__ float softplusf(float v) { return (v > 20.f) ? v : log1pf(__expf(v)); }

// Stage xc (post conv+silu) and optionally gate z for one chunk into LDS.
__device__ __forceinline__ void chunk_pointwise(
    const LayerParams& P, const float* __restrict__ Xin,
    int b, int S, int chunkStart, bool needZ,
    float* s_xcp0, float* s_xcp1, float* s_xc0, float* s_xc1,
    float* s_z0, float* s_z1)
{
  const int ln = threadIdx.x;
  const float nw0 = P.norm_w[0], nw1 = P.norm_w[1];
  const float* ip = P.in_proj;
  const float ip00 = ip[0], ip01 = ip[1], ip10 = ip[2], ip11 = ip[3];
  const float ip20 = ip[4], ip21 = ip[5], ip30 = ip[6], ip31 = ip[7];

  for (int i = ln; i < TCH + 3; i += BLK) {
    int tg = chunkStart - 3 + i;                 // global time (conv halo = 3)
    float x0 = 0.f, x1 = 0.f;
    if (tg >= 0 && tg < S) {
      const float* px = Xin + ((size_t)(b * S + tg)) * 2;
      x0 = px[0]; x1 = px[1];
    }
    float r   = rsqrtf(0.5f * (x0 * x0 + x1 * x1) + 1e-5f);   // rmsnorm, D_MODEL=2
    float xn0 = x0 * r * nw0, xn1 = x1 * r * nw1;
    s_xcp0[i] = ip00 * xn0 + ip01 * xn1;                      // pre-conv xc
    s_xcp1[i] = ip10 * xn0 + ip11 * xn1;
    if (needZ && i >= 3) {                                    // gate z
      s_z0[i - 3] = ip20 * xn0 + ip21 * xn1;
      s_z1[i - 3] = ip30 * xn0 + ip31 * xn1;
    }
  }
  __syncthreads();

  const float* cw = P.conv_w;
  const float cb0 = P.conv_b[0], cb1 = P.conv_b[1];
  for (int t = ln; t < TCH; t += BLK) {
    float a0 = cb0, a1 = cb1;
#pragma unroll
    for (int j = 0; j < 4; ++j) {                // causal depthwise conv, LDS idx t+j
      a0 += cw[j * 2 + 0] * s_xcp0[t + j];
      a1 += cw[j * 2 + 1] * s_xcp1[t + j];
    }
    s_xc0[t] = siluf(a0);
    s_xc1[t] = siluf(a1);
  }
  __syncthreads();
}

// Phase 1: local chunk scan -> carry pair (prod dA, local Bsum) per lane.
__global__ __launch_bounds__(BLK)
void mamba_phase1(LayerParams P, const float* __restrict__ Xin,
                  float* __restrict__ carryA, float* __restrict__ carryB,
                  int S, int nC)
{
  __shared__ float s_xcp0[TCH + 3], s_xcp1[TCH + 3], s_xc0[TCH], s_xc1[TCH];
  const int c = blockIdx.x, b = blockIdx.y, ln = threadIdx.x;
  const int chunkStart = c * TCH;
  chunk_pointwise(P, Xin, b, S, chunkStart, false,
                  s_xcp0, s_xcp1, s_xc0, s_xc1, nullptr, nullptr);

  if (ln < WAVE) {                               // wave0 only: serial lane scan
    const int e = ln >> 4, n = ln & 15;          // lane = (channel e, state n)
    const float* xp = P.x_proj;
    const float xpd0 = xp[0],           xpd1 = xp[1];                 // dt row
    const float xb0  = xp[(1 + n) * 2], xb1  = xp[(1 + n) * 2 + 1];   // B row n
    const float dtw  = P.dt_w[e],       dtb  = P.dt_b[e];
    const float Aen  = -__expf(P.A_log[e * NS_ + n]);

    float ca = 1.f, cb = 0.f;
    int tmax = S - chunkStart; if (tmax > TCH) tmax = TCH;
    for (int t = 0; t < tmax; ++t) {
      float x0 = s_xc0[t], x1 = s_xc1[t];
      float xce   = e ? x1 : x0;
      float dlt   = xpd0 * x0 + xpd1 * x1;
      float delta = softplusf(dtw * dlt + dtb);
      float Bv    = xb0 * x0 + xb1 * x1;
      float dA    = __expf(delta * Aen);
      ca *= dA;
      cb  = dA * cb + delta * Bv * xce;
    }
    const int idx = (b * nC + c) * WAVE + ln;
    carryA[idx] = ca;
    carryB[idx] = cb;
  }
}

// Phase 2: exclusive scan of chunk carries (serial over nC, trivial).
__global__ __launch_bounds__(WAVE)
void mamba_phase2(const float* __restrict__ cA, const float* __restrict__ cB,
                  float* __restrict__ pref, int nC)
{
  const int b = blockIdx.x, ln = threadIdx.x;
  float h = 0.f;
  for (int c = 0; c < nC; ++c) {
    const int idx = (b * nC + c) * WAVE + ln;
    pref[idx] = h;                               // h entering chunk c
    h = cA[idx] * h + cB[idx];
  }
}

// Phase 3: scan with prefix; stage h*C to LDS; reduce/gate/out_proj in parallel pass.
__global__ __launch_bounds__(BLK)
void mamba_phase3(LayerParams P, const float* __restrict__ Xin,
                  const float* __restrict__ prefix, float* __restrict__ Xout,
                  int S, int nC)
{
  __shared__ float s_xcp0[TCH + 3], s_xcp1[TCH + 3], s_xc0[TCH], s_xc1[TCH];
  __shared__ float s_z0[TCH], s_z1[TCH];
  __shared__ float s_ctr[TCH * CST];             // per-(t,lane) contrib, padded stride
  const int c = blockIdx.x, b = blockIdx.y, ln = threadIdx.x;
  const int chunkStart = c * TCH;
  chunk_pointwise(P, Xin, b, S, chunkStart, true,
                  s_xcp0, s_xcp1, s_xc0, s_xc1, s_z0, s_z1);

  int tmax = S - chunkStart; if (tmax > TCH) tmax = TCH;

  if (ln < WAVE) {                               // wave0: serial scan, NO shuffles
    const int e = ln >> 4, n = ln & 15;
    const float* xp = P.x_proj;
    const float xpd0 = xp[0],            xpd1 = xp[1];
    const float xb0  = xp[(1 + n) * 2],  xb1  = xp[(1 + n) * 2 + 1];
    const float cr0  = xp[(17 + n) * 2], cr1  = xp[(17 + n) * 2 + 1];  // C row n
    const float dtw  = P.dt_w[e],        dtb  = P.dt_b[e];
    const float Aen  = -__expf(P.A_log[e * NS_ + n]);

    float h = prefix[(b * nC + c) * WAVE + ln];
    for (int t = 0; t < tmax; ++t) {
      float x0 = s_xc0[t], x1 = s_xc1[t];
      float xce   = e ? x1 : x0;
      float dlt   = xpd0 * x0 + xpd1 * x1;
      float delta = softplusf(dtw * dlt + dtb);
      float Bv    = xb0 * x0 + xb1 * x1;
      float dA    = __expf(delta * Aen);
      h = dA * h + delta * Bv * xce;
      s_ctr[t * CST + ln] = h * (cr0 * x0 + cr1 * x1);  // fire-and-forget ds_store
    }
  }
  __syncthreads();

  // Parallel pass: reduce over n, skip-connection, gate, out_proj (both waves).
  const float D0 = P.Dp[0], D1 = P.Dp[1];
  const float* op = P.out_proj;
  const float op00 = op[0], op01 = op[1], op10 = op[2], op11 = op[3];
  for (int t = ln; t < tmax; t += BLK) {
    float s0 = 0.f, s1 = 0.f;
#pragma unroll
    for (int n = 0; n < NS_; ++n) {
      s0 += s_ctr[t * CST + n];
      s1 += s_ctr[t * CST + NS_ + n];
    }
    float y0 = (s0 + D0 * s_xc0[t]) * siluf(s_z0[t]);
    float y1 = (s1 + D1 * s_xc1[t]) * siluf(s_z1[t]);
    s_xcp0[t] = op00 * y0 + op01 * y1;           // stage residual delta (reuse xcp)
    s_xcp1[t] = op10 * y0 + op11 * y1;
  }
  __syncthreads();

  for (int t = ln; t < tmax; t += BLK) {         // coalesced residual write
    const size_t g = ((size_t)(b * S + chunkStart + t)) * 2;
    Xout[g]     = Xin[g]     + s_xcp0[t];
    Xout[g + 1] = Xin[g + 1] + s_xcp1[t];
  }
}

// Build mirrored input: X0[b,s] = z[b, s<L ? L-1-s : s-L], z = (x, dxdt*TAU/dt).
__global__ void build_input(const float* __restrict__ x, const float* __restrict__ dxdt,
                            const float* __restrict__ dtp, float* __restrict__ X0,
                            int L, int S)
{
  const float scale = TAU_ / dtp[0];
  int i = blockIdx.x * blockDim.x + threadIdx.x;      // over B*S
  if (i >= B_ * S) return;
  int b = i / S, s = i - b * S;
  int t = (s < L) ? (L - 1 - s) : (s - L);
  X0[(size_t)i * 2 + 0] = x[b * L + t];
  X0[(size_t)i * 2 + 1] = dxdt[b * L + t] * scale;
}

typedef __attribute__((ext_vector_type(2))) float v2f;
typedef __attribute__((ext_vector_type(8))) float v8f;

// Final heads. One wave = 16 timesteps x 16 kernels via V_WMMA_F32_16X16X4_F32
// (K = D_MODEL = 2, zero-padded to 4 in lanes 16..31 per the 32-bit A layout).
__global__ __launch_bounds__(256)
void final_kernel(const float* __restrict__ x, const float* __restrict__ dxdt,
                  const float* __restrict__ dtp,
                  const float* __restrict__ Xo, const float* __restrict__ Xg,
                  const float* __restrict__ Xk,
                  const float* __restrict__ onb, const float* __restrict__ onw,
                  const float* __restrict__ gnb, const float* __restrict__ gnw,
                  const float* __restrict__ kb,  const float* __restrict__ kcen,
                  const float* __restrict__ ksig, const float* __restrict__ kw,
                  float* __restrict__ yhat, float* __restrict__ wout,
                  int L, int S, int nRows)
{
  const int lane = threadIdx.x & 31;
  const int tile = blockIdx.x * (blockDim.x >> 5) + (threadIdx.x >> 5);
  const int row0 = tile * 16;                    // tile never crosses batch (L%16==0)
  if (row0 >= nRows) return;                     // wave-uniform; EXEC stays all-ones
  const int b  = row0 / L;
  const int t0 = row0 - b * L;
  const float scale = TAU_ / dtp[0];

  const int   lm  = lane & 15;
  const float inA = (lane < 16) ? 1.f : 0.f;     // K=2,3 zero-pad without EXEC masking
  // A (16x4, M=timestep, K=channel): vgpr0 = K0|K2, vgpr1 = K1|K3.
  const float* kc = Xk + ((size_t)(b * S + L + t0 + lm)) * 2;
  v2f a;  a.x  = kc[0] * inA;          a.y  = kc[1] * inA;
  // B (4x16, K=channel, N=kernel): row-per-half-wave, mirroring A.
  v2f bm; bm.x = kw[lm * 2 + 0] * inA; bm.y = kw[lm * 2 + 1] * inA;
  // C: bias broadcast, element (M, N=lm) = kb[lm] for every M.
  v8f cm;
  const float bias = kb[lm];
#pragma unroll
  for (int r = 0; r < 8; ++r) cm[r] = bias;

  v8f d = __builtin_amdgcn_wmma_f32_16x16x4_f32(
      /*neg_a=*/false, a, /*neg_b=*/false, bm,
      /*c_mod=*/(short)0, cm, /*reuse_a=*/false, /*reuse_b=*/false);

  const float invs = 1.f / (2.f * ksig[0] * ksig[0]);
  const float ow0 = onw[0], ow1 = onw[1], ob = onb[0];
  const float gw0 = gnw[0], gw1 = gnw[1], gb = gnb[0];
  const float c0 = kcen[lm * 2], c1 = kcen[lm * 2 + 1];
  const int half = (lane >> 4) << 3;             // D: vgpr r -> M = r + 8*(lane>=16)

#pragma unroll
  for (int r = 0; r < 8; ++r) {
    const int    t  = t0 + r + half;
    const size_t gi = (size_t)b * L + t;
    const float  w  = d[r];
    wout[gi * 16 + lm] = w;                      // weights output (B,L,16)

    const float z1 = x[gi];
    const float z2 = dxdt[gi] * scale;
    const float dd = (z1 - c0) * (z1 - c0) + (z2 - c1) * (z2 - c1);
    float contrib  = w * __expf(-dd * invs);     // weights * phi
    contrib += __shfl_xor(contrib, 1);
    contrib += __shfl_xor(contrib, 2);
    contrib += __shfl_xor(contrib, 4);
    contrib += __shfl_xor(contrib, 8);
    if (lm == 0) {
      const float* oc = Xo + ((size_t)(b * S + L + t)) * 2;
      const float* gc = Xg + ((size_t)(b * S + L + t)) * 2;
      const float om = fabsf(oc[0] * ow0 + oc[1] * ow1 + ob);
      const float gm = gc[0] * gw0 + gc[1] * gw1 + gb;
      yhat[gi] = -om * om * z1 - gm * z2 - contrib;
    }
  }
}

extern "C" void kernel_launch(void* const* d_in, const int* in_sizes, int n_in,
                              void* d_out, int out_size, void* d_ws, size_t ws_size,
                              hipStream_t stream)
{
  (void)n_in; (void)out_size; (void)ws_size;
  const float* x    = (const float*)d_in[0];
  const float* dxdt = (const float*)d_in[1];
  const float* dtp  = (const float*)d_in[2];
  const int L  = in_sizes[0] / B_;
  const int S  = 2 * L;
  const int nC = (S + TCH - 1) / TCH;

  // Workspace: ping/pong sequence buffers + per-mamba outputs + scan carries.
  float* ws = (float*)d_ws;
  const size_t seq = (size_t)B_ * S * 2;
  float* Xping  = ws;
  float* Xpong  = Xping + seq;
  float* Xfin[3] = { Xpong + seq, Xpong + 2 * seq, Xpong + 3 * seq };
  const size_t csz = (size_t)B_ * nC * WAVE;
  float* carryA = Xpong + 4 * seq;
  float* carryB = carryA + csz;
  float* prefix = carryB + csz;

  auto LP = [&](int m, int l) {
    const int base = 3 + m * 20 + l * 10;        // jax-pytree leaf order (alphabetical)
    LayerParams P;
    P.A_log   = (const float*)d_in[base + 0];
    P.Dp      = (const float*)d_in[base + 1];
    P.conv_b  = (const float*)d_in[base + 2];
    P.conv_w  = (const float*)d_in[base + 3];
    P.dt_b    = (const float*)d_in[base + 4];
    P.dt_w    = (const float*)d_in[base + 5];
    P.in_proj = (const float*)d_in[base + 6];
    P.norm_w  = (const float*)d_in[base + 7];
    P.out_proj= (const float*)d_in[base + 8];
    P.x_proj  = (const float*)d_in[base + 9];
    return P;
  };

  const int tot = B_ * S;
  build_input<<<(tot + 255) / 256, 256, 0, stream>>>(x, dxdt, dtp, Xping, L, S);

  const dim3 gScan(nC, B_);
  for (int m = 0; m < 3; ++m) {
    for (int l = 0; l < 2; ++l) {
      LayerParams P = LP(m, l);
      const float* Xi = (l == 0) ? Xping : Xpong;
      float*       Xo = (l == 0) ? Xpong : Xfin[m];
      mamba_phase1<<<gScan, BLK, 0, stream>>>(P, Xi, carryA, carryB, S, nC);
      mamba_phase2<<<B_, WAVE, 0, stream>>>(carryA, carryB, prefix, nC);
      mamba_phase3<<<gScan, BLK, 0, stream>>>(P, Xi, prefix, Xo, S, nC);
    }
  }

  float* yhat = (float*)d_out;
  float* wout = yhat + (size_t)B_ * L;           // (yhat, weights) concatenated
  const int nRows  = B_ * L;
  const int tiles  = nRows / 16;
  const int blocks = (tiles + 7) / 8;            // 8 waves / block
  final_kernel<<<blocks, 256, 0, stream>>>(
      x, dxdt, dtp, Xfin[0], Xfin[1], Xfin[2],
      (const float*)d_in[63], (const float*)d_in[64],   // omega_net: b, w
      (const float*)d_in[65], (const float*)d_in[66],   // gamma_net: b, w
      (const float*)d_in[67], (const float*)d_in[68],   // kernel: b, centers
      (const float*)d_in[69], (const float*)d_in[70],   // kernel: sigma, w
      yhat, wout, L, S, nRows);
}